// ScaledDotProductAttention_27479200760151
// MI455X (gfx1250) — compile-verified
//
#include <hip/hip_runtime.h>
#include <math.h>

typedef float v2f __attribute__((ext_vector_type(2)));
typedef float v8f __attribute__((ext_vector_type(8)));

constexpr int BATCH = 8;
constexpr int SEQ   = 2048;
constexpr int DIM   = 64;
constexpr int TILE  = 16;
constexpr int QPW   = 2;               // query tiles per wave
constexpr int KTILES = SEQ / TILE;     // 128
constexpr float SCALE = 0.125f;        // 1/sqrt(64)

// One wave32 handles two 16-query tiles of one batch, streaming all 2048 keys
// in 16-key tiles (flash attention, online softmax). K/V fragments are loaded
// once per key tile and feed two independent WMMA accumulation chains.
__global__ __launch_bounds__(32)
void ScaledDotProductAttention_fa_kernel(const float* __restrict__ q,
                                         const float* __restrict__ k,
                                         const float* __restrict__ v,
                                         float* __restrict__ out) {
    const int lane = threadIdx.x & 31;
    const int r = lane & 15;   // column within 16-lane half
    const int h = lane >> 4;   // half index
    const int qt = blockIdx.x * QPW;   // first query tile of this wave
    const int b  = blockIdx.y;         // batch

    const float* Qb = q + (size_t)b * SEQ * DIM;
    const float* K  = k + (size_t)b * SEQ * DIM;
    const float* V  = v + (size_t)b * SEQ * DIM;

    // Q fragments for both query tiles. For 16x16x4 f32 WMMA, the B-frag of
    // K*Q^T is {Q[r][4c+2h], Q[r][4c+2h+1]} per lane.
    v2f qf[QPW][16];
#pragma unroll
    for (int p = 0; p < QPW; ++p)
#pragma unroll
        for (int c = 0; c < 16; ++c)
            qf[p][c] = *(const v2f*)(Qb + (size_t)(qt + p) * TILE * DIM
                                        + r * DIM + 4 * c + 2 * h);

    // O accumulators: acc[p][n][e] = O_p[query = e+8h][d = 16n + r]
    v8f acc[QPW][4] = {};
    float mi[QPW] = {-INFINITY, -INFINITY};  // running max for query r
    float li[QPW] = {0.0f, 0.0f};            // running denom for query r

    for (int j = 0; j < KTILES; ++j) {
        const int kb = j * TILE;

        // S^T (16 keys x 16 queries) = K_tile x Q_p^T for both query tiles.
        // Two independent WMMA chains share each loaded K fragment.
        v8f st[QPW] = {};
#pragma unroll
        for (int c = 0; c < 16; ++c) {
            v2f kf = *(const v2f*)(K + (size_t)(kb + r) * DIM + 4 * c + 2 * h);
            st[0] = __builtin_amdgcn_wmma_f32_16x16x4_f32(false, kf, false, qf[0][c],
                                                          (short)0, st[0], false, false);
            st[1] = __builtin_amdgcn_wmma_f32_16x16x4_f32(false, kf, false, qf[1][c],
                                                          (short)0, st[1], false, false);
        }

        v2f pa[QPW][4];
#pragma unroll
        for (int p = 0; p < QPW; ++p) {
            // scale by 1/sqrt(d); equality mask: scaled score == 0 -> -inf
#pragma unroll
            for (int e = 0; e < 8; ++e) {
                float s = st[p][e] * SCALE;
                st[p][e] = (s == 0.0f) ? -INFINITY : s;
            }

            // online softmax for query r: 8 scores here + 8 in lane^16
            float tmax = st[p][0];
#pragma unroll
            for (int e = 1; e < 8; ++e) tmax = fmaxf(tmax, st[p][e]);
            tmax = fmaxf(tmax, __shfl_xor(tmax, 16, 32));
            float mnew = fmaxf(mi[p], tmax);
            float corr = __expf(mi[p] - mnew);   // exp(-inf)=0 on first tile
            mi[p] = mnew;

            float psum = 0.0f;
#pragma unroll
            for (int e = 0; e < 8; ++e) {
                float pe = __expf(st[p][e] - mnew);
                st[p][e] = pe;                   // st now holds P^T tile
                psum += pe;
            }
            psum += __shfl_xor(psum, 16, 32);
            li[p] = li[p] * corr + psum;

            // rescale O rows: acc[p][.][e] belongs to query e+8h
#pragma unroll
            for (int e = 0; e < 8; ++e) {
                float ce = __shfl(corr, e + 8 * h, 32);
                acc[p][0][e] *= ce; acc[p][1][e] *= ce;
                acc[p][2][e] *= ce; acc[p][3][e] *= ce;
            }

            // Build A-frags of P (16q x 16k) from P^T D-layout.
            // P[r][c] lives in st[c%8] of lane r + 16*(c/8).
            float sh[8];
#pragma unroll
            for (int e = 0; e < 8; ++e) sh[e] = __shfl_xor(st[p][e], 16, 32);
            pa[p][0][0] = h ? sh[2]    : st[p][0];  pa[p][0][1] = h ? sh[3]    : st[p][1];
            pa[p][1][0] = h ? sh[6]    : st[p][4];  pa[p][1][1] = h ? sh[7]    : st[p][5];
            pa[p][2][0] = h ? st[p][2] : sh[0];     pa[p][2][1] = h ? st[p][3] : sh[1];
            pa[p][3][0] = h ? st[p][6] : sh[4];     pa[p][3][1] = h ? st[p][7] : sh[5];
        }

        // O_p += P_p x V_tile ; each loaded V fragment feeds both chains.
#pragma unroll
        for (int n = 0; n < 4; ++n) {
#pragma unroll
            for (int kc = 0; kc < 4; ++kc) {
                const float* vp = V + (size_t)(kb + 4 * kc + 2 * h) * DIM + 16 * n + r;
                v2f bf;
                bf[0] = vp[0];
                bf[1] = vp[DIM];
                acc[0][n] = __builtin_amdgcn_wmma_f32_16x16x4_f32(false, pa[0][kc], false, bf,
                                                                  (short)0, acc[0][n], false, false);
                acc[1][n] = __builtin_amdgcn_wmma_f32_16x16x4_f32(false, pa[1][kc], false, bf,
                                                                  (short)0, acc[1][n], false, false);
            }
        }
    }

    // finalize: divide by softmax denom and store
#pragma unroll
    for (int p = 0; p < QPW; ++p) {
        float linv = 1.0f / li[p];
        float* O = out + (size_t)b * SEQ * DIM + (size_t)(qt + p) * TILE * DIM;
#pragma unroll
        for (int e = 0; e < 8; ++e) {
            float le = __shfl(linv, e + 8 * h, 32);
            const int row = e + 8 * h;
#pragma unroll
            for (int n = 0; n < 4; ++n)
                O[(size_t)row * DIM + 16 * n + r] = acc[p][n][e] * le;
        }
    }
}

extern "C" void kernel_launch(void* const* d_in, const int* in_sizes, int n_in,
                              void* d_out, int out_size, void* d_ws, size_t ws_size,
                              hipStream_t stream) {
    const float* q = (const float*)d_in[0];
    const float* k = (const float*)d_in[1];
    const float* v = (const float*)d_in[2];
    float* out = (float*)d_out;
    dim3 grid(SEQ / (TILE * QPW), BATCH);
    ScaledDotProductAttention_fa_kernel<<<grid, dim3(32), 0, stream>>>(q, k, v, out);
}